// ReprojectionLoss_90537910600253
// MI455X (gfx1250) — compile-verified
//
#include <hip/hip_runtime.h>
#include <cstdint>
#include <cstddef>

// ---------------- problem constants (from reference setup_inputs) ------------
namespace {
constexpr int kB = 8, kC = 3, kH = 256, kW = 832;
constexpr int TH = 8, TW = 64;          // interior tile
constexpr int SH = TH + 2, SW = TW + 2; // staged tile incl. reflect halo
constexpr int TILE = kC * SH * SW;      // 1980 floats per frame tile
constexpr int NTHREADS = TH * TW;       // 512 = 16 wave32s
constexpr int TILES_H = kH / TH;        // 32 (exact)
constexpr int TILES_W = kW / TW;        // 13 (exact)
constexpr int NBLOCKS = kB * TILES_H * TILES_W; // 3328
constexpr float kAlpha = 0.85f;
constexpr float kC1 = 1e-4f;   // 0.01^2
constexpr float kC2 = 9e-4f;   // 0.03^2
constexpr float kInv9 = 1.0f / 9.0f;
}

// ---------------- CDNA5 async global->LDS helpers ---------------------------
// GLOBAL_LOAD_ASYNC_TO_LDS_B32: VDST = per-lane LDS byte address,
// VADDR = per-lane 64-bit global address. Tracked by ASYNCcnt.
template <bool NT>
__device__ __forceinline__ void async_ld_b32(uint32_t lds_byte_addr, const float* g) {
    if constexpr (NT)
        asm volatile("global_load_async_to_lds_b32 %0, %1, off th:TH_LOAD_NT"
                     :: "v"(lds_byte_addr), "v"(g) : "memory");
    else
        asm volatile("global_load_async_to_lds_b32 %0, %1, off"
                     :: "v"(lds_byte_addr), "v"(g) : "memory");
}
__device__ __forceinline__ void wait_async0() {
    asm volatile("s_wait_asynccnt 0" ::: "memory");
}
// Flat address of an LDS object carries the LDS byte offset in its low 32 bits.
__device__ __forceinline__ uint32_t lds_addr(const void* p) {
    return (uint32_t)(uintptr_t)p;
}

__device__ __forceinline__ int reflect1(int i, int n) {
    i = (i < 0) ? -i : i;               // pad==1 -> only -1 and n occur
    return (i >= n) ? (2 * n - 2 - i) : i;
}

// Stage one reflect-padded 3-channel tile (async; caller waits + barriers).
template <bool NT>
__device__ __forceinline__ void stage_tile(float* lds, const float* img,
                                           int h0, int w0, int tid) {
#pragma unroll
    for (int it = 0; it < (TILE + NTHREADS - 1) / NTHREADS; ++it) {
        int e = tid + it * NTHREADS;
        if (e < TILE) {
            int ch  = e / (SH * SW);
            int rem = e - ch * (SH * SW);
            int r   = rem / SW;
            int c   = rem - r * SW;
            int gh  = reflect1(h0 - 1 + r, kH);
            int gw  = reflect1(w0 - 1 + c, kW);
            const float* src = img + ((size_t)ch * kH + gh) * (size_t)kW + gw;
            async_ld_b32<NT>(lds_addr(lds + e), src);
        }
    }
}

__device__ __forceinline__ const float* frame_base(int f, const float* prev,
                                                   const float* nextf,
                                                   const float* preds, int b) {
    const size_t img = (size_t)kC * kH * kW;
    const float* base = (f == 0) ? prev
                      : (f == 1) ? nextf
                      : preds + (size_t)(f - 2) * (size_t)kB * img;
    return base + (size_t)b * img;
}

// ---------------- main fused kernel -----------------------------------------
// Block = (TW, TH) threads; one thread per interior pixel.
__global__ __launch_bounds__(NTHREADS)
void reproj_main(const float* __restrict__ gt, const float* __restrict__ prev,
                 const float* __restrict__ nextf, const float* __restrict__ preds,
                 const float* __restrict__ noise, float* __restrict__ block_sums) {
    __shared__ float gt_t[TILE];
    __shared__ float fr_t[2][TILE];
    __shared__ float wred[NTHREADS / 32];

    const int c   = threadIdx.x;            // 0..63  (w within tile)
    const int r   = threadIdx.y;            // 0..7   (h within tile)
    const int tid = r * TW + c;
    const int b   = blockIdx.z;
    const int h0  = blockIdx.y * TH, w0 = blockIdx.x * TW;
    const int h   = h0 + r, w = w0 + c;
    const size_t img = (size_t)kC * kH * kW;

    // Stage gt tile + first comparison tile together, then wait once.
    // gt/prev/next keep default (RT) caching; preds stream is marked NT below.
    stage_tile<false>(gt_t, gt + (size_t)b * img, h0, w0, tid);
    stage_tile<false>(fr_t[0], frame_base(0, prev, nextf, preds, b), h0, w0, tid);
    wait_async0();
    __syncthreads();

    // gt 3x3 window -> registers, and all gt-only SSIM terms hoisted out of
    // the 10-frame loop: mx, mx^2, 2*mx, vx (reused 10x each).
    float gwin[3][9], gcen[3], mxc[3], mx2[3], two_mx[3], vxc[3];
#pragma unroll
    for (int ch = 0; ch < 3; ++ch) {
        float sx = 0.f, sxx = 0.f;
#pragma unroll
        for (int dy = 0; dy < 3; ++dy)
#pragma unroll
            for (int dx = 0; dx < 3; ++dx) {
                float v = gt_t[ch * SH * SW + (r + dy) * SW + (c + dx)];
                gwin[ch][dy * 3 + dx] = v;
                sx += v; sxx += v * v;
            }
        float mx = sx * kInv9;
        mxc[ch]    = mx;
        mx2[ch]    = mx * mx;
        two_mx[ch] = 2.f * mx;
        vxc[ch]    = sxx * kInv9 - mx * mx;
        gcen[ch]   = gwin[ch][4];
    }

    // Photometric for: f=0 prev, f=1 next, f=2..9 preds[s][fr] (k=f-2=s*2+fr)
    float ph[10];
#pragma unroll
    for (int f = 0; f < 10; ++f) {
        const int cur = f & 1;
        // Double buffer: prefetch next frame's tile while computing this one.
        // preds (204 MB) is strictly streaming -> non-temporal staging.
        if (f == 0)
            stage_tile<false>(fr_t[1], frame_base(1, prev, nextf, preds, b),
                              h0, w0, tid);
        else if (f < 9)
            stage_tile<true>(fr_t[cur ^ 1], frame_base(f + 1, prev, nextf, preds, b),
                             h0, w0, tid);

        const float* ft = fr_t[cur];
        float ssim_sum = 0.f, l1_sum = 0.f;
#pragma unroll
        for (int ch = 0; ch < 3; ++ch) {
            float sy = 0.f, syy = 0.f, sxy = 0.f, ycen = 0.f;
#pragma unroll
            for (int dy = 0; dy < 3; ++dy)
#pragma unroll
                for (int dx = 0; dx < 3; ++dx) {
                    float yv = ft[ch * SH * SW + (r + dy) * SW + (c + dx)];
                    float xv = gwin[ch][dy * 3 + dx];
                    sy += yv; syy += yv * yv; sxy += xv * yv;
                    if (dy == 1 && dx == 1) ycen = yv;
                }
            float my  = sy * kInv9;
            float vy  = syy * kInv9 - my * my;
            float cov = sxy * kInv9 - mxc[ch] * my;
            float num = (two_mx[ch] * my + kC1) * (2.f * cov + kC2);
            float den = (mx2[ch] + my * my + kC1) * (vxc[ch] + vy + kC2);
            float ssim = num / den;
            ssim = fminf(fmaxf(ssim, 0.f), 1.f);   // clip per channel, like ref
            ssim_sum += ssim;
            l1_sum   += fabsf(gcen[ch] - ycen);
        }
        ph[f] = (1.f - kAlpha) * (l1_sum * (1.f / 3.f))
              + kAlpha * 0.5f * (1.f - ssim_sum * (1.f / 3.f));

        wait_async0();     // prefetched tile landed (this wave)
        __syncthreads();   // all waves done reading cur + all tiles landed
    }

    // min over {recon[s,0], recon[s,1], motion0+eps*n, motion1+eps*n}, sum scales
    const size_t hw  = (size_t)kH * kW;
    const size_t pix = (size_t)h * kW + w;
    float total = 0.f;
#pragma unroll
    for (int s = 0; s < 4; ++s) {
        // noise (68 MB) is streaming -> non-temporal loads
        float n0 = __builtin_nontemporal_load(
                       noise + ((size_t)(s * 2 + 0) * kB + b) * hw + pix);
        float n1 = __builtin_nontemporal_load(
                       noise + ((size_t)(s * 2 + 1) * kB + b) * hw + pix);
        float m0 = ph[0] + 1e-5f * n0;
        float m1 = ph[1] + 1e-5f * n1;
        float r0 = ph[2 + s * 2 + 0];
        float r1 = ph[2 + s * 2 + 1];
        total += fminf(fminf(r0, r1), fminf(m0, m1));
    }

    // deterministic block reduction: wave32 shfl tree, then cross-wave tree
    float v = total;
#pragma unroll
    for (int off = 16; off > 0; off >>= 1) v += __shfl_down(v, off, 32);
    const int lane = tid & 31, wid = tid >> 5;
    if (lane == 0) wred[wid] = v;
    __syncthreads();
    if (wid == 0) {
        float t = wred[lane & (NTHREADS / 32 - 1)];
#pragma unroll
        for (int off = (NTHREADS / 64); off > 0; off >>= 1)
            t += __shfl_down(t, off, 32);
        if (lane == 0) {
            int bid = ((int)blockIdx.z * gridDim.y + blockIdx.y) * gridDim.x
                    + blockIdx.x;
            block_sums[bid] = t;
        }
    }
}

// ---------------- finalize: deterministic sum of block partials --------------
__global__ __launch_bounds__(256)
void reproj_finalize(const float* __restrict__ block_sums, float* __restrict__ out,
                     int n, float inv_count) {
    __shared__ float red[256];
    float s = 0.f;
    for (int i = threadIdx.x; i < n; i += 256) s += block_sums[i];
    red[threadIdx.x] = s;
    __syncthreads();
    for (int off = 128; off > 0; off >>= 1) {
        if ((int)threadIdx.x < off) red[threadIdx.x] += red[threadIdx.x + off];
        __syncthreads();
    }
    if (threadIdx.x == 0) out[0] = red[0] * inv_count;
}

// ---------------- host launcher ---------------------------------------------
extern "C" void kernel_launch(void* const* d_in, const int* in_sizes, int n_in,
                              void* d_out, int out_size, void* d_ws, size_t ws_size,
                              hipStream_t stream) {
    const float* gt    = (const float*)d_in[0];
    const float* prev  = (const float*)d_in[1];
    const float* nextf = (const float*)d_in[2];
    const float* preds = (const float*)d_in[3];
    const float* noise = (const float*)d_in[4];
    float* out = (float*)d_out;
    float* block_sums = (float*)d_ws;   // NBLOCKS floats, fully rewritten each call

    dim3 grid(TILES_W, TILES_H, kB);
    dim3 block(TW, TH);
    hipLaunchKernelGGL(reproj_main, grid, block, 0, stream,
                       gt, prev, nextf, preds, noise, block_sums);

    const float inv_count = 1.0f / (float)((size_t)kB * kH * kW);
    hipLaunchKernelGGL(reproj_finalize, dim3(1), dim3(256), 0, stream,
                       block_sums, out, NBLOCKS, inv_count);
}